// PointNet2_73083163509342
// MI455X (gfx1250) — compile-verified
//
#include <hip/hip_runtime.h>
#include <hip/hip_bf16.h>

// ---------------------------------------------------------------------------
// PointNet++ (SSG) forward for gfx1250 (MI455X), wave32.
// GEMMs: V_WMMA_F32_16X16X4_F32. Block = 8 waves stages a 64x64 X chunk in
// LDS once; the 8 waves cover 8 O-tiles (128 cols), killing the O/16-fold
// HBM re-read of the activation matrix (the real roofline limiter here).
// ---------------------------------------------------------------------------

typedef __attribute__((ext_vector_type(2))) float v2f;
typedef __attribute__((ext_vector_type(8))) float v8f;

#define EPSBN 1e-5f
#define LDS_STRIDE 68   // 64 + 4: conflict-free banks for 16-row x 4-K frag reads

// ---------------------------------------------------------------------------
// (4,32,3,2048) -> [128][2048][3]
__global__ void transpose_xyz_k(const float* __restrict__ in, float* __restrict__ out) {
    long i = (long)blockIdx.x * blockDim.x + threadIdx.x;
    const long total = 128L * 2048L * 3L;
    if (i >= total) return;
    int b = (int)(i / (2048 * 3));
    int r = (int)(i % (2048 * 3));
    int n = r / 3, k = r % 3;
    out[i] = in[((long)b * 3 + k) * 2048 + n];
}

// ---------------------------------------------------------------------------
// Farthest point sampling: one block (256 thr) per batch. Matches jax scan:
// emits `far` BEFORE the distance update; argmax ties -> lowest index.
__global__ void fps_k(const float* __restrict__ xyzt, int N, int S, int* __restrict__ fidx) {
    const int b = blockIdx.x;
    const int t = threadIdx.x;              // 256 threads
    __shared__ float rd[256];
    __shared__ int   ri[256];
    const int per = N >> 8;                 // N/256 (2048->8, 512->2)
    float dl[8];
    for (int j = 0; j < per; ++j) dl[j] = 1e10f;
    const float* base = xyzt + (long)b * N * 3;
    int far = 0;
    for (int s = 0; s < S; ++s) {
        if (t == 0) fidx[b * S + s] = far;
        float cx = base[far * 3 + 0], cy = base[far * 3 + 1], cz = base[far * 3 + 2];
        float bd = -1.0f; int bi = 0;
        for (int j = 0; j < per; ++j) {
            int i = t + (j << 8);
            float dx = base[i * 3 + 0] - cx;
            float dy = base[i * 3 + 1] - cy;
            float dz = base[i * 3 + 2] - cz;
            float d = dx * dx + dy * dy + dz * dz;
            if (d < dl[j]) dl[j] = d;
            if (dl[j] > bd) { bd = dl[j]; bi = i; }   // ascending i => lowest idx on tie
        }
        rd[t] = bd; ri[t] = bi;
        __syncthreads();
        for (int off = 128; off > 0; off >>= 1) {
            if (t < off) {
                float d2 = rd[t + off]; int i2 = ri[t + off];
                if (d2 > rd[t] || (d2 == rd[t] && i2 < ri[t])) { rd[t] = d2; ri[t] = i2; }
            }
            __syncthreads();
        }
        far = ri[0];
        __syncthreads();
    }
}

// ---------------------------------------------------------------------------
// Ball query: one wave32 per query; keeps first `ns` in-radius points in index
// order (== reference sort-by-index), pads with the first valid index.
// Also writes new_xyz = xyzt[fidx].
__global__ void ballq_k(const float* __restrict__ xyzt, const int* __restrict__ fidx,
                        int B, int N, int S, int ns, float r2,
                        float* __restrict__ newxyz, int* __restrict__ idx) {
    const int lane = threadIdx.x & 31;
    const long q = ((long)blockIdx.x * blockDim.x + threadIdx.x) >> 5;
    if (q >= (long)B * S) return;
    const int b = (int)(q / S);
    const int fid = fidx[q];
    const float* base = xyzt + (long)b * N * 3;
    const float cx = base[fid * 3 + 0], cy = base[fid * 3 + 1], cz = base[fid * 3 + 2];
    if (lane == 0) { newxyz[q * 3 + 0] = cx; newxyz[q * 3 + 1] = cy; newxyz[q * 3 + 2] = cz; }
    int cnt = 0, first = -1;
    for (int bj = 0; bj < N && cnt < ns; bj += 32) {
        int j = bj + lane;
        float dx = base[j * 3 + 0] - cx;
        float dy = base[j * 3 + 1] - cy;
        float dz = base[j * 3 + 2] - cz;
        bool in = (dx * dx + dy * dy + dz * dz) <= r2;
        unsigned long long mask = __ballot((int)in);
        if (first < 0 && mask) first = bj + (__ffsll((unsigned long long)mask) - 1);
        int pos = cnt + (int)__popcll(mask & ((1ull << lane) - 1ull));
        if (in && pos < ns) idx[q * ns + pos] = j;
        cnt += (int)__popcll(mask);
    }
    if (first < 0) first = 0;
    for (int p = cnt + lane; p < ns; p += 32) idx[q * ns + p] = first;
}

// ---------------------------------------------------------------------------
// Gather grouped features, one thread per OUTPUT ELEMENT (coalesced over C):
// X[row=(b,s,n)][0:3] = xyz[idx]-center, [3:3+Cp] = pts[idx]
__global__ void group_k(const float* __restrict__ xyzt, const float* __restrict__ pts,
                        const float* __restrict__ newxyz, const int* __restrict__ idx,
                        int B, int Npts, int S, int ns, int Cp, float* __restrict__ X) {
    const int C = 3 + Cp;
    long e = (long)blockIdx.x * blockDim.x + threadIdx.x;
    long total = (long)B * S * ns * C;
    if (e >= total) return;
    long row = e / C;
    int c = (int)(e % C);
    int n = (int)(row % ns);
    long q = row / ns;                       // b*S + s
    int b = (int)(q / S);
    int j = idx[q * ns + n];
    float v;
    if (c < 3) {
        v = xyzt[((long)b * Npts + j) * 3 + c] - newxyz[q * 3 + c];
    } else {
        v = pts[((long)b * Npts + j) * Cp + (c - 3)];
    }
    X[e] = v;
}

// group_all: X[row][0:3]=xyz[row], [3:]=pts[row]  (no centering)
__global__ void groupall_k(const float* __restrict__ xyzt, const float* __restrict__ pts,
                           long rows, int Cp, float* __restrict__ X) {
    const int C = 3 + Cp;
    long e = (long)blockIdx.x * blockDim.x + threadIdx.x;
    if (e >= rows * (long)C) return;
    long row = e / C;
    int c = (int)(e % C);
    X[e] = (c < 3) ? xyzt[row * 3 + c] : pts[row * (long)Cp + (c - 3)];
}

// ---------------------------------------------------------------------------
// Fused Y = act((X . W^T) * scale + shift).  X:[M,K] row-major, W:[O,K].
// Block (8 waves): stages X chunk [64 rows x 64 K] in LDS once; wave w owns
// O-tile (blockO*8 + w). Each wave computes a 64x16 tile = 4 sub-tiles.
// A frags come from LDS (stride-68 rows: conflict-free, 8B aligned);
// B frags from global (weights are small and cache-hot).
// K tail: LDS chunk is zero-filled on the A side; B loads in the tail are
// BRANCHLESS (clamped index + v_cndmask select) since for K=3 the whole GEMM
// runs the tail path. Out-of-range rows/cols clamped; stores guarded.
__global__ void wmma_gemm_bn_relu(const float* __restrict__ X, int M, int K,
                                  const float* __restrict__ W, int O,
                                  const float* __restrict__ bias,
                                  const float* __restrict__ g,
                                  const float* __restrict__ beta,
                                  const float* __restrict__ mean,
                                  const float* __restrict__ var,
                                  int has_bn, int do_relu,
                                  float* __restrict__ Y, int ldY) {
    __shared__ float tile[64 * LDS_STRIDE];

    const int  lane = threadIdx.x & 31;
    const int  wave = threadIdx.x >> 5;            // 0..7
    const int  oT   = (O + 15) >> 4;
    const int  oB   = (oT + 7) >> 3;               // O-tile groups of 8
    const long tS   = (long)blockIdx.x / oB;       // 64-row super-tile
    const int  tOb  = (int)((long)blockIdx.x % oB);

    // A frag (16x4 f32): lanes 0-15 -> K={k,k+1}, lanes 16-31 -> K={k+2,k+3}, M=lane&15
    // B frag (4x16 f32): same K split, N=lane&15
    const int  rsel  = lane & 15;
    const int  khalf = (lane >> 4) << 1;           // 0 or 2
    const int  tO    = tOb * 8 + wave;
    const int  col   = tO * 16 + rsel;
    const bool cOK   = (col < O);
    const int  colC  = cOK ? col : (O - 1);        // clamp (idle waves / fc3)
    const float* wrow = W + (long)colC * K;

    const long row0 = tS * 64;

    v8f acc[4] = {};

    for (int k0 = 0; k0 < K; k0 += 64) {
        const int kc = (K - k0 < 64) ? (K - k0) : 64;
        // ---- stage X[row0:row0+64, k0:k0+64] into LDS (zero-fill K tail) ----
        __syncthreads();
#pragma unroll
        for (int i = 0; i < 16; ++i) {
            int e   = (int)threadIdx.x + (i << 8);     // 0..4095
            int r   = e >> 6;                          // 0..63
            int c   = e & 63;                          // 0..63
            long gr = row0 + r;
            if (gr > (long)M - 1) gr = (long)M - 1;    // clamp rows
            int cc  = (c < kc) ? c : (kc - 1);         // clamp cols (branchless)
            float v = X[gr * (long)K + k0 + cc];
            tile[r * LDS_STRIDE + c] = (c < kc) ? v : 0.0f;
        }
        __syncthreads();
        // ---- compute on the chunk ----
        const float* wr = wrow + k0;
        if (kc == 64) {                                 // hot path, no guards
#pragma unroll 4
            for (int step = 0; step < 16; ++step) {
                const int kk = (step << 2) + khalf;
                v2f bf;
                bf.x = wr[kk];
                bf.y = wr[kk + 1];
                v2f af[4];
#pragma unroll
                for (int i = 0; i < 4; ++i) {
                    const int a = (rsel + (i << 4)) * LDS_STRIDE + kk;
                    af[i].x = tile[a];
                    af[i].y = tile[a + 1];
                }
#pragma unroll
                for (int i = 0; i < 4; ++i)
                    acc[i] = __builtin_amdgcn_wmma_f32_16x16x4_f32(
                        false, af[i], false, bf, (short)0, acc[i], false, false);
            }
        } else {                                        // partial chunk (K=3: only path)
            const int steps = (kc + 3) >> 2;
            for (int step = 0; step < steps; ++step) {
                const int kk = (step << 2) + khalf;
                // branchless: clamped loads (always in-bounds) + register select
                const int k0c = (kk     < kc) ? kk     : (kc - 1);
                const int k1c = (kk + 1 < kc) ? kk + 1 : (kc - 1);
                float b0 = wr[k0c];
                float b1 = wr[k1c];
                v2f bf;
                bf.x = (kk     < kc) ? b0 : 0.0f;
                bf.y = (kk + 1 < kc) ? b1 : 0.0f;
                v2f af[4];
#pragma unroll
                for (int i = 0; i < 4; ++i) {           // LDS zero-filled: unguarded
                    const int a = (rsel + (i << 4)) * LDS_STRIDE + kk;
                    af[i].x = tile[a];
                    af[i].y = tile[a + 1];
                }
#pragma unroll
                for (int i = 0; i < 4; ++i)
                    acc[i] = __builtin_amdgcn_wmma_f32_16x16x4_f32(
                        false, af[i], false, bf, (short)0, acc[i], false, false);
            }
        }
    }

    // ---- epilogue: BN + ReLU + store ----
    float scale = 1.0f, shift = 0.0f;
    if (has_bn) {
        scale = g[colC] * rsqrtf(var[colC] + EPSBN);
        shift = (bias[colC] - mean[colC]) * scale + beta[colC];
    } else if (bias) {
        shift = bias[colC];
    }
    // D frag: VGPR r holds M = base + r + 8*(lane>>4), N = lane&15
    const int mhi = (lane >> 4) << 3;
#pragma unroll
    for (int i = 0; i < 4; ++i) {
        const long mbase = row0 + 16 * i + mhi;
#pragma unroll
        for (int r = 0; r < 8; ++r) {
            long m = mbase + r;
            if (m < (long)M && cOK) {
                float v = acc[i][r] * scale + shift;
                if (do_relu) v = fmaxf(v, 0.0f);
                Y[m * (long)ldY + col] = v;
            }
        }
    }
}

// ---------------------------------------------------------------------------
__global__ void maxpool_k(const float* __restrict__ Y, long Bs, int ns, int O,
                          float* __restrict__ out) {
    long i = (long)blockIdx.x * blockDim.x + threadIdx.x;
    if (i >= Bs * (long)O) return;
    long bs = i / O; int o = (int)(i % O);
    const float* p = Y + (bs * ns) * (long)O + o;
    float m = -3.402823466e38f;
    for (int n = 0; n < ns; ++n) m = fmaxf(m, p[(long)n * O]);
    out[bs * (long)O + o] = m;
}

__global__ void copy_k(const float* __restrict__ src, float* __restrict__ dst, long n) {
    long i = (long)blockIdx.x * blockDim.x + threadIdx.x;
    if (i < n) dst[i] = src[i];
}

// log_softmax over 14 classes, write pred[b][c][t] (B=4,t=32)
__global__ void logsoftmax_pred_k(const float* __restrict__ logits, float* __restrict__ pred) {
    int row = blockIdx.x * blockDim.x + threadIdx.x;   // 0..127
    if (row >= 128) return;
    float v[14];
    float mx = -3.402823466e38f;
    for (int c = 0; c < 14; ++c) { v[c] = logits[row * 14 + c]; mx = fmaxf(mx, v[c]); }
    float s = 0.0f;
    for (int c = 0; c < 14; ++c) s += expf(v[c] - mx);
    float lse = mx + logf(s);
    int b = row >> 5, t = row & 31;
    for (int c = 0; c < 14; ++c) pred[((long)b * 14 + c) * 32 + t] = v[c] - lse;
}

// ---------------------------------------------------------------------------
static inline void launch_gemm(const float* X, int M, int K, const float* W, int O,
                               const float* bias, const float* g, const float* beta,
                               const float* mean, const float* var,
                               int has_bn, int relu, float* Y, int ldY, hipStream_t s) {
    int  oT = (O + 15) / 16;
    int  oB = (oT + 7) / 8;
    long mS = ((long)M + 63) / 64;
    unsigned grid = (unsigned)(mS * oB);
    wmma_gemm_bn_relu<<<grid, 256, 0, s>>>(X, M, K, W, O, bias, g, beta, mean, var,
                                           has_bn, relu, Y, ldY);
}

extern "C" void kernel_launch(void* const* d_in, const int* in_sizes, int n_in,
                              void* d_out, int out_size, void* d_ws, size_t ws_size,
                              hipStream_t stream) {
    (void)in_sizes; (void)n_in; (void)out_size; (void)ws_size;
    // ------- inputs (flattened dict order) -------
    const float* xyz = (const float*)d_in[0];
    auto SA = [&](int sa, int layer, int item) -> const float* {
        return (const float*)d_in[1 + sa * 18 + layer * 6 + item];
    };
    const float* fc1_W = (const float*)d_in[55];
    const float* fc1_b = (const float*)d_in[56];
    const float* fc2_W = (const float*)d_in[57];
    const float* fc2_b = (const float*)d_in[58];
    const float* fc3_W = (const float*)d_in[59];
    const float* fc3_b = (const float*)d_in[60];
    const float* bn1g = (const float*)d_in[61], *bn1b = (const float*)d_in[62];
    const float* bn1m = (const float*)d_in[63], *bn1v = (const float*)d_in[64];
    const float* bn2g = (const float*)d_in[65], *bn2b = (const float*)d_in[66];
    const float* bn2m = (const float*)d_in[67], *bn2v = (const float*)d_in[68];
    float* out = (float*)d_out;

    // ------- workspace layout (float offsets) -------
    float* ws = (float*)d_ws;
    float* XYZT  = ws;                              // 786432
    int*   FIDX1 = (int*)(ws + 786432);             // 65536
    float* L1X   = ws + 851968;                     // 196608
    int*   IDX1  = (int*)(ws + 1048576);            // 2097152
    float* L1P   = ws + 3145728;                    // 8388608
    int*   FIDX2 = (int*)(ws + 11534336);           // 16384
    float* L2X   = ws + 11550720;                   // 49152
    int*   IDX2  = (int*)(ws + 11599872);           // 1048576
    float* L2P   = ws + 12648448;                   // 4194304
    float* L3P   = ws + 16842752;                   // 131072
    float* H1    = ws + 16973824;                   // 65536
    float* H2    = ws + 17039360;                   // 32768
    float* LG    = ws + 17072128;                   // 2048
    float* BA    = ws + 17074176;                   // 268435456
    float* BB    = ws + 285509632;                  // 268435456

    const int B = 128;

    // ------- stage 0: transpose -------
    transpose_xyz_k<<<(786432 + 255) / 256, 256, 0, stream>>>(xyz, XYZT);

    // ------- SA1: npoint=512, r=0.2, ns=32, 3 -> 64 -> 64 -> 128 -------
    fps_k<<<B, 256, 0, stream>>>(XYZT, 2048, 512, FIDX1);
    {   long q = (long)B * 512;                               // 65536 queries, 8 waves/blk
        ballq_k<<<(unsigned)((q + 7) / 8), 256, 0, stream>>>(XYZT, FIDX1, B, 2048, 512,
                                                             32, 0.04f, L1X, IDX1);
    }
    {   long rows = (long)B * 512 * 32;                       // 2,097,152
        long el = rows * 3;
        group_k<<<(unsigned)((el + 255) / 256), 256, 0, stream>>>(
            XYZT, nullptr, L1X, IDX1, B, 2048, 512, 32, 0, BA);
        int M = (int)rows;
        launch_gemm(BA, M, 3,  SA(0,0,0), 64,  SA(0,0,1), SA(0,0,2), SA(0,0,3), SA(0,0,4), SA(0,0,5), 1, 1, BB, 64,  stream);
        launch_gemm(BB, M, 64, SA(0,1,0), 64,  SA(0,1,1), SA(0,1,2), SA(0,1,3), SA(0,1,4), SA(0,1,5), 1, 1, BA, 64,  stream);
        launch_gemm(BA, M, 64, SA(0,2,0), 128, SA(0,2,1), SA(0,2,2), SA(0,2,3), SA(0,2,4), SA(0,2,5), 1, 1, BB, 128, stream);
        long po = (long)B * 512 * 128;
        maxpool_k<<<(unsigned)((po + 255) / 256), 256, 0, stream>>>(BB, (long)B * 512, 32, 128, L1P);
    }

    // ------- SA2: npoint=128, r=0.4, ns=64, 131 -> 128 -> 128 -> 256 -------
    fps_k<<<B, 256, 0, stream>>>(L1X, 512, 128, FIDX2);
    {   long q = (long)B * 128;
        ballq_k<<<(unsigned)((q + 7) / 8), 256, 0, stream>>>(L1X, FIDX2, B, 512, 128,
                                                             64, 0.16f, L2X, IDX2);
    }
    {   long rows = (long)B * 128 * 64;                       // 1,048,576
        long el = rows * 131;
        group_k<<<(unsigned)((el + 255) / 256), 256, 0, stream>>>(
            L1X, L1P, L2X, IDX2, B, 512, 128, 64, 128, BA);
        int M = (int)rows;
        launch_gemm(BA, M, 131, SA(1,0,0), 128, SA(1,0,1), SA(1,0,2), SA(1,0,3), SA(1,0,4), SA(1,0,5), 1, 1, BB, 128, stream);
        launch_gemm(BB, M, 128, SA(1,1,0), 128, SA(1,1,1), SA(1,1,2), SA(1,1,3), SA(1,1,4), SA(1,1,5), 1, 1, BA, 128, stream);
        launch_gemm(BA, M, 128, SA(1,2,0), 256, SA(1,2,1), SA(1,2,2), SA(1,2,3), SA(1,2,4), SA(1,2,5), 1, 1, BB, 256, stream);
        long po = (long)B * 128 * 256;
        maxpool_k<<<(unsigned)((po + 255) / 256), 256, 0, stream>>>(BB, (long)B * 128, 64, 256, L2P);
    }

    // ------- SA3 (group_all): 259 -> 256 -> 512 -> 1024, maxpool over 128 -------
    {   long rows = (long)B * 128;                            // 16384
        long el = rows * 259;
        groupall_k<<<(unsigned)((el + 255) / 256), 256, 0, stream>>>(L2X, L2P, rows, 256, BA);
        int M = (int)rows;
        launch_gemm(BA, M, 259, SA(2,0,0), 256,  SA(2,0,1), SA(2,0,2), SA(2,0,3), SA(2,0,4), SA(2,0,5), 1, 1, BB, 256,  stream);
        launch_gemm(BB, M, 256, SA(2,1,0), 512,  SA(2,1,1), SA(2,1,2), SA(2,1,3), SA(2,1,4), SA(2,1,5), 1, 1, BA, 512,  stream);
        launch_gemm(BA, M, 512, SA(2,2,0), 1024, SA(2,2,1), SA(2,2,2), SA(2,2,3), SA(2,2,4), SA(2,2,5), 1, 1, BB, 1024, stream);
        long po = (long)B * 1024;
        maxpool_k<<<(unsigned)((po + 255) / 256), 256, 0, stream>>>(BB, (long)B, 128, 1024, L3P);
    }

    // second output: l3p reshaped (4,32,1024) == row-major [128][1024]
    copy_k<<<(131072 + 255) / 256, 256, 0, stream>>>(L3P, out + 1792, 131072);

    // ------- FC head -------
    launch_gemm(L3P, 128, 1024, fc1_W, 512, fc1_b, bn1g, bn1b, bn1m, bn1v, 1, 1, H1, 512, stream);
    launch_gemm(H1,  128, 512,  fc2_W, 256, fc2_b, bn2g, bn2b, bn2m, bn2v, 1, 1, H2, 256, stream);
    launch_gemm(H2,  128, 256,  fc3_W, 14,  fc3_b, nullptr, nullptr, nullptr, nullptr, 0, 0, LG, 14, stream);

    logsoftmax_pred_k<<<1, 128, 0, stream>>>(LG, out);
}